// FireSpreadGAT_2637109920210
// MI455X (gfx1250) — compile-verified
//
#include <hip/hip_runtime.h>
#include <stdint.h>

// ---------------- problem constants (from reference) ----------------
#define NNODES 50000
#define NEDGES 800000
#define ETOT   (NEDGES + NNODES)   // edges + self loops = 850000
#define FIN    128
#define NH     4
#define CH     32
#define HC     (NH * CH)           // 128

// ---------------- CDNA5 types ----------------
typedef float v2f __attribute__((ext_vector_type(2)));
typedef float v8f __attribute__((ext_vector_type(8)));
typedef unsigned int u32x4 __attribute__((ext_vector_type(4)));
typedef int i32x8 __attribute__((ext_vector_type(8)));
typedef int i32x4 __attribute__((ext_vector_type(4)));

// monotone float <-> u32 key (so atomicMax(u32) == float max; 0 is below -inf)
__device__ __forceinline__ unsigned fkey(float f) {
  int i = __float_as_int(f);
  return (i >= 0) ? ((unsigned)i | 0x80000000u) : ~(unsigned)i;
}
__device__ __forceinline__ float unkey(unsigned k) {
  int i = (k & 0x80000000u) ? (int)(k & 0x7FFFFFFFu) : ~(int)k;
  return __int_as_float(i);
}
__device__ __forceinline__ float lrelu(float v) { return v > 0.f ? v : 0.2f * v; }

// ---------------- TDM: async 2-D fp32 tile load, global(row-major) -> LDS ----
// D# per CDNA5 ISA. pad_enable: +2 dwords per 128 dwords so
// the LDS tile has row stride 130 floats (bank-conflict-free A-fragment reads).
// This toolchain exposes the 6-arg clang-23 builtin:
//   (uint32x4 g0, int32x8 g1, int32x4 g2, int32x4 g3, int32x8 extra, i32 cpol)
__device__ __forceinline__ void tdm_load_tile(const float* g, unsigned ldsByte,
                                              unsigned tileW, unsigned tileH,
                                              unsigned tensorW, unsigned tensorH)
{
  unsigned long long ga = (unsigned long long)(uintptr_t)g;
  u32x4 g0;
  g0[0] = 1u;                                                  // count = 1
  g0[1] = ldsByte;                                             // LDS dest (bytes)
  g0[2] = (unsigned)ga;                                        // global addr lo
  g0[3] = (unsigned)((ga >> 32) & 0x01FFFFFFu) | (2u << 30);   // addr[56:32] | type=2
  i32x8 g1;
  //       data_size=4B     pad_en      interval=128dw  amount=2dw
  g1[0] = (int)((2u << 16) | (1u << 20) | (6u << 22) | (1u << 25));
  g1[1] = (int)((tensorW & 0xFFFFu) << 16);                    // tensor_dim0 lo16
  g1[2] = (int)(((tensorW >> 16) & 0xFFFFu) | ((tensorH & 0xFFFFu) << 16));
  g1[3] = (int)(((tensorH >> 16) & 0xFFFFu) | ((tileW & 0xFFFFu) << 16));
  g1[4] = (int)(tileH & 0xFFFFu);                              // tile_dim1, tile_dim2=0
  g1[5] = (int)tensorW;                                        // dim0 stride (elems)
  g1[6] = 0;
  g1[7] = 0;
  i32x4 gz4 = {0, 0, 0, 0};
  i32x8 gz8 = {0, 0, 0, 0, 0, 0, 0, 0};
  __builtin_amdgcn_tensor_load_to_lds(g0, g1, gz4, gz4, gz8, 0);
}

// ---------------- fp32 WMMA GEMM: D[N x NCOL] = A[N x 128] * B[128 x NCOL] ----
// one 16-row tile per block, one 16-col tile per wave, K swept 4 at a time with
// V_WMMA_F32_16X16X4_F32 (exact fp32; GEMM is far from any roof here).
#define LDSROW (FIN + 2)   // 130 floats: matches TDM pad

template<int NCOL>
__global__ void __launch_bounds__(NCOL * 2)
gemm_wmma(const float* __restrict__ A, const float* __restrict__ B,
          float* __restrict__ D)
{
  __shared__ float sA[16 * LDSROW];
  const int row0 = blockIdx.x * 16;
  const int wave = threadIdx.x >> 5;
  const int lane = threadIdx.x & 31;

  if (threadIdx.x < 32) {   // wave 0 drives the DMA, then drains TENSORcnt
    tdm_load_tile(A + (size_t)row0 * FIN, (unsigned)(uintptr_t)(&sA[0]),
                  FIN, 16, FIN, NNODES);
    __builtin_amdgcn_s_wait_tensorcnt(0);
  }
  __syncthreads();

  const int col0 = wave * 16;
  const int mn   = lane & 15;        // A row / B col for this lane
  const int kh   = (lane >> 4) * 2;  // lanes 16..31 hold K+2,K+3
  v8f acc = {};
#pragma unroll
  for (int k = 0; k < FIN; k += 4) {
    const int k0 = k + kh;
    v2f a, b;
    a.x = sA[mn * LDSROW + k0];
    a.y = sA[mn * LDSROW + k0 + 1];
    b.x = B[k0 * NCOL + col0 + mn];
    b.y = B[(k0 + 1) * NCOL + col0 + mn];
    acc = __builtin_amdgcn_wmma_f32_16x16x4_f32(false, a, false, b,
                                                (short)0, acc, false, false);
  }
  // D layout: lanes 0-15 -> M = r, lanes 16-31 -> M = r + 8
  float* dp = D + (size_t)row0 * NCOL + col0 + mn;
  const int mbase = (lane >> 4) * 8;
#pragma unroll
  for (int r = 0; r < 8; ++r)
    dp[(size_t)(mbase + r) * NCOL] = acc[r];
}

// ---------------- attention logits per node ----------------
template<int H>
__global__ void att_dots(const float* __restrict__ hfeat,
                         const float* __restrict__ asrc,
                         const float* __restrict__ adst,
                         float* __restrict__ es, float* __restrict__ ed)
{
  int t = blockIdx.x * blockDim.x + threadIdx.x;
  if (t >= NNODES * H) return;
  int n = t / H, h = t % H;
  const float* hp = hfeat + (size_t)n * (H * CH) + h * CH;
  const float* ap = asrc + h * CH;
  const float* dp = adst + h * CH;
  float s1 = 0.f, s2 = 0.f;
#pragma unroll
  for (int c = 0; c < CH; ++c) { float v = hp[c]; s1 += v * ap[c]; s2 += v * dp[c]; }
  es[t] = s1; ed[t] = s2;
}

// ---------------- segment softmax passes over edges ----------------
__device__ __forceinline__ void edge_sd(int e, const int* ei, int& s, int& d) {
  if (e < NEDGES) { s = ei[e]; d = ei[NEDGES + e]; } else { s = d = e - NEDGES; }
}

template<int H>
__global__ void edge_max(const int* __restrict__ ei, const float* __restrict__ es,
                         const float* __restrict__ ed, unsigned* __restrict__ mx)
{
  int e = blockIdx.x * blockDim.x + threadIdx.x;
  if (e >= ETOT) return;
  int s, d; edge_sd(e, ei, s, d);
#pragma unroll
  for (int h = 0; h < H; ++h)
    atomicMax(mx + d * H + h, fkey(lrelu(es[s * H + h] + ed[d * H + h])));
}

template<int H>
__global__ void edge_expsum(const int* __restrict__ ei, const float* __restrict__ es,
                            const float* __restrict__ ed, const unsigned* __restrict__ mx,
                            float* __restrict__ den)
{
  int e = blockIdx.x * blockDim.x + threadIdx.x;
  if (e >= ETOT) return;
  int s, d; edge_sd(e, ei, s, d);
#pragma unroll
  for (int h = 0; h < H; ++h) {
    float v = lrelu(es[s * H + h] + ed[d * H + h]);
    atomicAdd(den + d * H + h, __expf(v - unkey(mx[d * H + h])));
  }
}

// layer-1 aggregation: one wave per edge, 4 channels/lane (head = lane>>3)
__global__ void aggregate1(const int* __restrict__ ei, const float* __restrict__ h1,
                           const float* __restrict__ es, const float* __restrict__ ed,
                           const unsigned* __restrict__ mx, const float* __restrict__ den,
                           float* __restrict__ out)
{
  int e    = (blockIdx.x * blockDim.x + threadIdx.x) >> 5;
  int lane = threadIdx.x & 31;
  if (e >= ETOT) return;
  int s, d; edge_sd(e, ei, s, d);
  int h = lane >> 3;
  float v = lrelu(es[s * NH + h] + ed[d * NH + h]);
  float alpha = __expf(v - unkey(mx[d * NH + h])) / (den[d * NH + h] + 1e-16f);
  const float4 hv = *(const float4*)(h1 + (size_t)s * HC + lane * 4);
  float* op = out + (size_t)d * HC + lane * 4;
  atomicAdd(op + 0, hv.x * alpha);
  atomicAdd(op + 1, hv.y * alpha);
  atomicAdd(op + 2, hv.z * alpha);
  atomicAdd(op + 3, hv.w * alpha);
}

// layer-2 aggregation: one wave per edge, 1 channel/lane (single head)
__global__ void aggregate2(const int* __restrict__ ei, const float* __restrict__ h2,
                           const float* __restrict__ es, const float* __restrict__ ed,
                           const unsigned* __restrict__ mx, const float* __restrict__ den,
                           float* __restrict__ out)
{
  int e    = (blockIdx.x * blockDim.x + threadIdx.x) >> 5;
  int lane = threadIdx.x & 31;
  if (e >= ETOT) return;
  int s, d; edge_sd(e, ei, s, d);
  float v = lrelu(es[s] + ed[d]);
  float alpha = __expf(v - unkey(mx[d])) / (den[d] + 1e-16f);
  atomicAdd(out + (size_t)d * CH + lane, h2[(size_t)s * CH + lane] * alpha);
}

__global__ void bias_relu(const float* __restrict__ in, const float* __restrict__ b,
                          float* __restrict__ out, int total)
{
  int i = blockIdx.x * blockDim.x + threadIdx.x;
  if (i >= total) return;
  float v = in[i] + b[i & (HC - 1)];
  out[i] = v > 0.f ? v : 0.f;
}

__global__ void finalize(const float* __restrict__ agg, const float* __restrict__ b2,
                         const float* __restrict__ fw, const float* __restrict__ fb,
                         float* __restrict__ y)
{
  int n = blockIdx.x * blockDim.x + threadIdx.x;
  if (n >= NNODES) return;
  float z = 0.f;
#pragma unroll
  for (int c = 0; c < CH; ++c) {
    float v = agg[(size_t)n * CH + c] + b2[c];
    v = v > 0.f ? v : 0.f;
    z += v * fw[c];
  }
  z += fb[0];
  y[n] = 1.f / (1.f + __expf(-z));
}

// ---------------- host-side orchestration ----------------
extern "C" void kernel_launch(void* const* d_in, const int* in_sizes, int n_in,
                              void* d_out, int out_size, void* d_ws, size_t ws_size,
                              hipStream_t stream)
{
  (void)in_sizes; (void)n_in; (void)out_size; (void)ws_size;
  const float* x   = (const float*)d_in[0];
  const int*   ei  = (const int*)d_in[1];
  const float* W1  = (const float*)d_in[2];
  const float* as1 = (const float*)d_in[3];
  const float* ad1 = (const float*)d_in[4];
  const float* b1  = (const float*)d_in[5];
  const float* W2  = (const float*)d_in[6];
  const float* as2 = (const float*)d_in[7];
  const float* ad2 = (const float*)d_in[8];
  const float* b2  = (const float*)d_in[9];
  const float* fw  = (const float*)d_in[10];
  const float* fb  = (const float*)d_in[11];
  float* y = (float*)d_out;

  // workspace layout (floats/u32, ~68 MB total)
  float*    h1   = (float*)d_ws;                        // N*128
  float*    es1  = h1   + (size_t)NNODES * HC;          // N*4
  float*    ed1  = es1  + (size_t)NNODES * NH;          // N*4
  unsigned* mx1  = (unsigned*)(ed1 + (size_t)NNODES * NH); // N*4   (zero region 1 start)
  float*    den1 = (float*)mx1 + (size_t)NNODES * NH;   // N*4
  float*    out1 = den1 + (size_t)NNODES * NH;          // N*128
  float*    h2p  = out1 + (size_t)NNODES * HC;          // N*32
  float*    es2  = h2p  + (size_t)NNODES * CH;          // N
  float*    ed2  = es2  + NNODES;                       // N
  unsigned* mx2  = (unsigned*)(ed2 + NNODES);           // N     (zero region 2 start)
  float*    den2 = (float*)mx2 + NNODES;                // N
  float*    out2 = den2 + NNODES;                       // N*32

  // per-call zero init (graph-capture safe). fkey encoding => 0 < key(-inf).
  (void)hipMemsetAsync(mx1, 0, (size_t)(2 * NNODES * NH + NNODES * HC) * 4, stream);
  (void)hipMemsetAsync(mx2, 0, (size_t)(2 * NNODES + NNODES * CH) * 4, stream);

  const int TB = 256;
  const int egrid = (ETOT + TB - 1) / TB;
  const int wgrid = (ETOT * 32 + TB - 1) / TB;   // wave-per-edge kernels

  // ---- layer 1 ----
  gemm_wmma<HC><<<NNODES / 16, 256, 0, stream>>>(x, W1, h1);
  att_dots<NH><<<(NNODES * NH + TB - 1) / TB, TB, 0, stream>>>(h1, as1, ad1, es1, ed1);
  edge_max<NH><<<egrid, TB, 0, stream>>>(ei, es1, ed1, mx1);
  edge_expsum<NH><<<egrid, TB, 0, stream>>>(ei, es1, ed1, mx1, den1);
  aggregate1<<<wgrid, TB, 0, stream>>>(ei, h1, es1, ed1, mx1, den1, out1);
  bias_relu<<<(NNODES * HC + TB - 1) / TB, TB, 0, stream>>>(out1, b1, h1, NNODES * HC);

  // ---- layer 2 ----
  gemm_wmma<CH><<<NNODES / 16, 64, 0, stream>>>(h1, W2, h2p);
  att_dots<1><<<(NNODES + TB - 1) / TB, TB, 0, stream>>>(h2p, as2, ad2, es2, ed2);
  edge_max<1><<<egrid, TB, 0, stream>>>(ei, es2, ed2, mx2);
  edge_expsum<1><<<egrid, TB, 0, stream>>>(ei, es2, ed2, mx2, den2);
  aggregate2<<<wgrid, TB, 0, stream>>>(ei, h2p, es2, ed2, mx2, den2, out2);

  // ---- head ----
  finalize<<<(NNODES + TB - 1) / TB, TB, 0, stream>>>(out2, b2, fw, fb, y);
}